// ApplyFullFisheyeWarping_33741263077739
// MI455X (gfx1250) — compile-verified
//
#include <hip/hip_runtime.h>

// Fisheye warp + bilinear resample, MI455X (gfx1250).
// Bandwidth-bound gather: ~75.5 MB stores + ~100-150 MB gathered reads
// -> ~8-10 us at 23.3 TB/s. No matmul structure -> no WMMA (would be fake
// work). Strategy: one thread per output pixel, warp coords computed once,
// reused across 8 planes (grid.z covers the 96 B*C planes), fully unrolled
// so 32 global_load_b32 are in flight per thread. NT stores keep the 192MB
// L2 for the gathered input; global_prefetch_b8 warms L2 for the next
// plane group (consumed by sibling blocks).

namespace {
constexpr int kH = 960;
constexpr int kW = 1280;
constexpr int kB = 32;
constexpr int kC = 3;
constexpr int kCH = 288;              // int(H * 0.3)
constexpr int kCW = 384;              // int(W * 0.3)
constexpr int kOH = kH - 2 * kCH;     // 384
constexpr int kOW = kW - 2 * kCW;     // 512
constexpr int kPlanes = kB * kC;      // 96
constexpr int kPL = 8;                // planes handled per thread
constexpr int kHW = kH * kW;
constexpr int kOHW = kOH * kOW;
constexpr int kBlockX = 128;          // 4 wave32 per block
}  // namespace

__global__ __launch_bounds__(kBlockX) void fisheye_warp_kernel(
    const float* __restrict__ img,    // (96, 960, 1280)
    const float* __restrict__ Kmat,   // (3,3) row-major
    const float* __restrict__ dist,   // (4,)
    float* __restrict__ out)          // (96, 384, 512)
{
  const int ox = blockIdx.x * kBlockX + threadIdx.x;  // 0..511
  const int oy = blockIdx.y;                          // 0..383
  const int p0 = blockIdx.z * kPL;                    // plane group base

  // Uniform camera params (compiler keeps these scalar).
  const float fx = Kmat[0], cx = Kmat[2];
  const float fy = Kmat[4], cy = Kmat[5];
  const float k1 = dist[0], k2 = dist[1], k3 = dist[2], k4 = dist[3];

  // --- warp coordinates: computed once, reused for kPL planes ---
  const float xn = ((float)(ox + kCW) - cx) / fx;
  const float yn = ((float)(oy + kCH) - cy) / fy;
  const float r2 = xn * xn + yn * yn;
  const float dfac = 1.0f + r2 * (k1 + r2 * (k2 + r2 * (k3 + r2 * k4)));
  const float xd = xn * dfac * fx + cx;
  const float yd = yn * dfac * fy + cy;

  const float ix0f = floorf(xd);
  const float iy0f = floorf(yd);
  const float wx1 = xd - ix0f, wy1 = yd - iy0f;
  const float wx0 = 1.0f - wx1, wy0 = 1.0f - wy1;

  const int ix0 = (int)ix0f, iy0 = (int)iy0f;
  const int ix1 = ix0 + 1,   iy1 = iy0 + 1;

  // Validity masks folded into the weights (reference semantics).
  const float vx0 = (ix0 >= 0 && ix0 < kW) ? 1.0f : 0.0f;
  const float vx1 = (ix1 >= 0 && ix1 < kW) ? 1.0f : 0.0f;
  const float vy0 = (iy0 >= 0 && iy0 < kH) ? 1.0f : 0.0f;
  const float vy1 = (iy1 >= 0 && iy1 < kH) ? 1.0f : 0.0f;

  const float w00 = wy0 * wx0 * vy0 * vx0;
  const float w01 = wy0 * wx1 * vy0 * vx1;
  const float w10 = wy1 * wx0 * vy1 * vx0;
  const float w11 = wy1 * wx1 * vy1 * vx1;

  const int ix0c = min(max(ix0, 0), kW - 1);
  const int ix1c = min(max(ix1, 0), kW - 1);
  const int iy0c = min(max(iy0, 0), kH - 1);
  const int iy1c = min(max(iy1, 0), kH - 1);

  const int o00 = iy0c * kW + ix0c;
  const int o01 = iy0c * kW + ix1c;
  const int o10 = iy1c * kW + ix0c;
  const int o11 = iy1c * kW + ix1c;

  const float* __restrict__ ip = img + (size_t)p0 * kHW;
  float* __restrict__ op = out + (size_t)p0 * kOHW + (size_t)oy * kOW + ox;

  // Warm L2 (DEV scope) for the next plane group; consumed by sibling
  // blocks -> speculative global_prefetch_b8 into the shared 192MB L2.
  if (p0 + kPL < kPlanes) {
    __builtin_prefetch(ip + (size_t)kPL * kHW + o00, 0, 1);
    __builtin_prefetch(ip + (size_t)kPL * kHW + o10, 0, 1);
  }

  // Fully unrolled: 32 independent gather loads in flight, then FMAs,
  // then 8 streamed NT stores (written once, never re-read).
#pragma unroll
  for (int p = 0; p < kPL; ++p) {
    const float* __restrict__ q = ip + (size_t)p * kHW;
    const float v = q[o00] * w00 + q[o01] * w01 +
                    q[o10] * w10 + q[o11] * w11;
    __builtin_nontemporal_store(v, op + (size_t)p * kOHW);
  }
}

extern "C" void kernel_launch(void* const* d_in, const int* in_sizes, int n_in,
                              void* d_out, int out_size, void* d_ws, size_t ws_size,
                              hipStream_t stream) {
  const float* img  = (const float*)d_in[0];   // (32,3,960,1280) fp32
  const float* Kmat = (const float*)d_in[1];   // (3,3) fp32
  const float* dist = (const float*)d_in[2];   // (4,) fp32
  float* out = (float*)d_out;                  // (32,3,384,512) fp32

  dim3 block(kBlockX, 1, 1);
  dim3 grid(kOW / kBlockX,          // 4
            kOH,                    // 384
            kPlanes / kPL);         // 12
  fisheye_warp_kernel<<<grid, block, 0, stream>>>(img, Kmat, dist, out);
}